// GATFeatureExtractor_55353538511234
// MI455X (gfx1250) — compile-verified
//
#include <hip/hip_runtime.h>
#include <hip/hip_bf16.h>
#include <cstdint>
#include <cstddef>

#define N_NODES 20000
#define N_EDGES 320000
#define EPRIME  (N_EDGES + N_NODES)   // 340000 (with self loops)
#define N_GRAPH 64
#define IN_CH   128
#define HID     64
#define HEADS   8
#define F1      (HEADS * HID)         // 512
#define LEAKY   0.2f
#define KEY_NEG_INF 0x007FFFFFu       // fkey(-inf)

typedef float v2f __attribute__((ext_vector_type(2)));
typedef float v8f __attribute__((ext_vector_type(8)));

// ---- ordered-key float atomic max helpers -------------------------------
__device__ __forceinline__ unsigned fkey(float f) {
  unsigned b = __float_as_uint(f);
  return (b & 0x80000000u) ? ~b : (b | 0x80000000u);
}
__device__ __forceinline__ float funkey(unsigned k) {
  unsigned b = (k & 0x80000000u) ? (k & 0x7FFFFFFFu) : ~k;
  return __uint_as_float(b);
}

// ---- fills --------------------------------------------------------------
__global__ void k_fill_f32(float* p, float v, size_t n) {
  size_t i = (size_t)blockIdx.x * blockDim.x + threadIdx.x;
  size_t s = (size_t)gridDim.x * blockDim.x;
  for (; i < n; i += s) p[i] = v;
}
__global__ void k_fill_u32(unsigned* p, unsigned v, size_t n) {
  size_t i = (size_t)blockIdx.x * blockDim.x + threadIdx.x;
  size_t s = (size_t)gridDim.x * blockDim.x;
  for (; i < n; i += s) p[i] = v;
}

// ---- WMMA fp32 GEMM:  C[n,m] = sum_k A[n,k] * B[m,k]  (A·B^T) -----------
// One wave computes a 16x64 strip of C (4 column tiles). The A fragment is
// loaded once per K-step and reused across 4 WMMAs -> 4x less A traffic.
// nColTiles must be a multiple of 4; row count a multiple of 16.
__global__ void k_wmma_gemm_nt4(const float* __restrict__ A,
                                const float* __restrict__ B,
                                float* __restrict__ C,
                                int nRowTiles, int nColGroups, int K, int ldc) {
  int gid  = blockIdx.x * blockDim.x + threadIdx.x;
  int wave = gid >> 5;
  int lane = threadIdx.x & 31;
  int total = nRowTiles * nColGroups;
  if (wave >= total) return;                 // uniform per-wave -> EXEC all 1s inside
  int rt  = wave / nColGroups;
  int ctg = wave % nColGroups;

  int idx   = lane & 15;                     // row (A) / col (B) within tile
  int khalf = (lane >> 4) << 1;              // 0 or 2
  const float* ap = A + (size_t)(rt * 16 + idx) * K + khalf;
  const float* bp0 = B + (size_t)(ctg * 64 + 0  + idx) * K + khalf;
  const float* bp1 = B + (size_t)(ctg * 64 + 16 + idx) * K + khalf;
  const float* bp2 = B + (size_t)(ctg * 64 + 32 + idx) * K + khalf;
  const float* bp3 = B + (size_t)(ctg * 64 + 48 + idx) * K + khalf;

  v8f acc0 = {}, acc1 = {}, acc2 = {}, acc3 = {};
  for (int k = 0; k < K; k += 4) {
    v2f a  = *reinterpret_cast<const v2f*>(ap  + k);
    v2f b0 = *reinterpret_cast<const v2f*>(bp0 + k);
    v2f b1 = *reinterpret_cast<const v2f*>(bp1 + k);
    v2f b2 = *reinterpret_cast<const v2f*>(bp2 + k);
    v2f b3 = *reinterpret_cast<const v2f*>(bp3 + k);
    acc0 = __builtin_amdgcn_wmma_f32_16x16x4_f32(false, a, false, b0, (short)0, acc0, false, false);
    acc1 = __builtin_amdgcn_wmma_f32_16x16x4_f32(false, a, false, b1, (short)0, acc1, false, false);
    acc2 = __builtin_amdgcn_wmma_f32_16x16x4_f32(false, a, false, b2, (short)0, acc2, false, false);
    acc3 = __builtin_amdgcn_wmma_f32_16x16x4_f32(false, a, false, b3, (short)0, acc3, false, false);
  }

  int rowb = rt * 16 + ((lane >> 4) << 3);   // +8 for upper half-wave
  int colb = ctg * 64 + idx;
#pragma unroll
  for (int v = 0; v < 8; ++v) {
    float* cp = C + (size_t)(rowb + v) * ldc + colb;
    cp[0]  = acc0[v];
    cp[16] = acc1[v];
    cp[32] = acc2[v];
    cp[48] = acc3[v];
  }
}

// ---- per-node attention dots: a_s[n,h], a_d[n,h] ------------------------
__global__ void k_node_attn(const float* __restrict__ h,
                            const float* __restrict__ att_src,
                            const float* __restrict__ att_dst,
                            float* __restrict__ as, float* __restrict__ ad,
                            int n, int H, int C) {
  int idx = blockIdx.x * blockDim.x + threadIdx.x;
  if (idx >= n * H) return;
  int node = idx / H, hh = idx % H;
  const float* hp = h + ((size_t)node * H + hh) * C;
  const float* sp = att_src + hh * C;
  const float* dp = att_dst + hh * C;
  float vs = 0.f, vd = 0.f;
  for (int c = 0; c < C; ++c) { float x = hp[c]; vs += x * sp[c]; vd += x * dp[c]; }
  as[idx] = vs; ad[idx] = vd;
}

// ---- edge logits + segment max ------------------------------------------
__global__ void k_edge_logit_max(const int* __restrict__ ei,
                                 const float* __restrict__ as,
                                 const float* __restrict__ ad,
                                 float* __restrict__ eatt,
                                 unsigned* __restrict__ m, int H) {
  int idx = blockIdx.x * blockDim.x + threadIdx.x;
  if (idx >= EPRIME * H) return;
  int e = idx / H, hh = idx % H;
  int s, d;
  if (e < N_EDGES) { s = ei[e]; d = ei[N_EDGES + e]; } else { s = d = e - N_EDGES; }
  float v = as[s * H + hh] + ad[d * H + hh];
  v = v > 0.f ? v : LEAKY * v;
  eatt[idx] = v;
  atomicMax(&m[d * H + hh], fkey(v));
}

// ---- exp(e - max) + segment sum -----------------------------------------
__global__ void k_edge_exp_sum(const int* __restrict__ ei,
                               float* __restrict__ eatt,
                               const unsigned* __restrict__ m,
                               float* __restrict__ z, int H) {
  int idx = blockIdx.x * blockDim.x + threadIdx.x;
  if (idx >= EPRIME * H) return;
  int e = idx / H, hh = idx % H;
  int d = (e < N_EDGES) ? ei[N_EDGES + e] : e - N_EDGES;
  float v = __expf(eatt[idx] - funkey(m[d * H + hh]));
  eatt[idx] = v;
  atomicAdd(&z[d * H + hh], v);
}

// ---- normalized weighted message scatter-add ----------------------------
__global__ void k_edge_scatter(const int* __restrict__ ei,
                               const float* __restrict__ eatt,
                               const float* __restrict__ z,
                               const float* __restrict__ h,
                               float* __restrict__ out, int H, int C) {
  int q = C >> 2;  // float4 groups per head
  long long idx = (long long)blockIdx.x * blockDim.x + threadIdx.x;
  long long total = (long long)EPRIME * H * q;
  if (idx >= total) return;
  int e  = (int)(idx / (H * q));
  int r  = (int)(idx % (H * q));
  int hh = r / q;
  int c4 = (r % q) << 2;
  int s, d;
  if (e < N_EDGES) { s = ei[e]; d = ei[N_EDGES + e]; } else { s = d = e - N_EDGES; }
  float a = eatt[(size_t)e * H + hh] / (z[(size_t)d * H + hh] + 1e-16f);
  const float4 hv = *reinterpret_cast<const float4*>(h + ((size_t)s * H + hh) * C + c4);
  float* op = out + ((size_t)d * H + hh) * C + c4;
  atomicAdd(op + 0, hv.x * a);
  atomicAdd(op + 1, hv.y * a);
  atomicAdd(op + 2, hv.z * a);
  atomicAdd(op + 3, hv.w * a);
}

// ---- relu(x + b) in place + pooled sums ---------------------------------
__global__ void k_relu_bias_pool(float* __restrict__ xm, const float* __restrict__ b,
                                 const int* __restrict__ batch,
                                 float* __restrict__ pool, int n, int F) {
  int i = blockIdx.x * blockDim.x + threadIdx.x;
  if (i >= n * F) return;
  int node = i / F, f = i % F;
  float v = xm[i] + b[f];
  v = v > 0.f ? v : 0.f;
  xm[i] = v;
  atomicAdd(&pool[(size_t)batch[node] * F + f], v);
}

__global__ void k_count(const int* __restrict__ batch, float* __restrict__ cnt, int n) {
  int i = blockIdx.x * blockDim.x + threadIdx.x;
  if (i < n) atomicAdd(&cnt[batch[i]], 1.0f);
}

__global__ void k_pool_final(const float* __restrict__ pool, const float* __restrict__ cnt,
                             float* __restrict__ out, int total, int F) {
  int i = blockIdx.x * blockDim.x + threadIdx.x;
  if (i < total) out[i] = pool[i] / fmaxf(cnt[i / F], 1.0f);
}

__global__ void k_add_bias(float* __restrict__ x, const float* __restrict__ b,
                           int total, int F) {
  int i = blockIdx.x * blockDim.x + threadIdx.x;
  if (i < total) x[i] += b[i % F];
}

// -------------------------------------------------------------------------
static inline unsigned cdivu(long long a, long long b) { return (unsigned)((a + b - 1) / b); }

extern "C" void kernel_launch(void* const* d_in, const int* in_sizes, int n_in,
                              void* d_out_, int out_size, void* d_ws, size_t ws_size,
                              hipStream_t stream) {
  const float* x        = (const float*)d_in[0];
  const int*   ei       = (const int*)d_in[1];
  const int*   batch    = (const int*)d_in[2];
  const float* W1       = (const float*)d_in[3];
  const float* att_src1 = (const float*)d_in[4];
  const float* att_dst1 = (const float*)d_in[5];
  const float* b1       = (const float*)d_in[6];
  const float* W2       = (const float*)d_in[7];
  const float* att_src2 = (const float*)d_in[8];
  const float* att_dst2 = (const float*)d_in[9];
  const float* b2       = (const float*)d_in[10];
  float* d_out = (float*)d_out_;

  // workspace carve-up (256B aligned)
  char* ws = (char*)d_ws;
  size_t off = 0;
  auto carve = [&](size_t bytes) -> void* {
    void* p = ws + off;
    off += (bytes + 255) & ~(size_t)255;
    return p;
  };
  float*    h1   = (float*)carve((size_t)N_NODES * F1 * 4);
  float*    out1 = (float*)carve((size_t)N_NODES * F1 * 4);       // becomes x_mid
  float*    as1  = (float*)carve((size_t)N_NODES * HEADS * 4);
  float*    ad1  = (float*)carve((size_t)N_NODES * HEADS * 4);
  unsigned* m1   = (unsigned*)carve((size_t)N_NODES * HEADS * 4);
  float*    z1   = (float*)carve((size_t)N_NODES * HEADS * 4);
  float*    eatt = (float*)carve((size_t)EPRIME * HEADS * 4);     // reused by layer 2
  float*    pool = (float*)carve((size_t)N_GRAPH * F1 * 4);
  float*    cnt  = (float*)carve((size_t)N_GRAPH * 4);
  float*    h2   = (float*)carve((size_t)N_NODES * HID * 4);
  float*    as2  = (float*)carve((size_t)N_NODES * 4);
  float*    ad2  = (float*)carve((size_t)N_NODES * 4);
  unsigned* m2   = (unsigned*)carve((size_t)N_NODES * 4);
  float*    z2   = (float*)carve((size_t)N_NODES * 4);
  float*    xlate = d_out + (size_t)N_GRAPH * F1;                 // layer-2 output region

  const dim3 B(256);

  // ---- init (every call: harness does not re-zero) ----
  k_fill_f32<<<2048, B, 0, stream>>>(out1, 0.f, (size_t)N_NODES * F1);
  k_fill_f32<<<256,  B, 0, stream>>>(z1, 0.f, (size_t)N_NODES * HEADS);
  k_fill_u32<<<256,  B, 0, stream>>>(m1, KEY_NEG_INF, (size_t)N_NODES * HEADS);
  k_fill_f32<<<64,   B, 0, stream>>>(pool, 0.f, (size_t)N_GRAPH * F1);
  k_fill_f32<<<1,    B, 0, stream>>>(cnt, 0.f, (size_t)N_GRAPH);
  k_fill_f32<<<2048, B, 0, stream>>>(d_out, 0.f, (size_t)out_size);
  k_fill_f32<<<128,  B, 0, stream>>>(z2, 0.f, (size_t)N_NODES);
  k_fill_u32<<<128,  B, 0, stream>>>(m2, KEY_NEG_INF, (size_t)N_NODES);

  // ---- layer 1: h1 = x @ W1^T  (20000x128 · 128x512) ----
  {
    int waves = (N_NODES / 16) * (F1 / 64);           // 1250 * 8
    k_wmma_gemm_nt4<<<cdivu(waves, 8), B, 0, stream>>>(x, W1, h1,
                                                       N_NODES / 16, F1 / 64, IN_CH, F1);
  }
  k_node_attn<<<cdivu((long long)N_NODES * HEADS, 256), B, 0, stream>>>(
      h1, att_src1, att_dst1, as1, ad1, N_NODES, HEADS, HID);
  k_edge_logit_max<<<cdivu((long long)EPRIME * HEADS, 256), B, 0, stream>>>(
      ei, as1, ad1, eatt, m1, HEADS);
  k_edge_exp_sum<<<cdivu((long long)EPRIME * HEADS, 256), B, 0, stream>>>(
      ei, eatt, m1, z1, HEADS);
  k_edge_scatter<<<cdivu((long long)EPRIME * HEADS * (HID / 4), 256), B, 0, stream>>>(
      ei, eatt, z1, h1, out1, HEADS, HID);

  // ---- relu + bias + mean pool -> d_out[0 : 64*512] ----
  k_relu_bias_pool<<<cdivu((long long)N_NODES * F1, 256), B, 0, stream>>>(
      out1, b1, batch, pool, N_NODES, F1);
  k_count<<<cdivu(N_NODES, 256), B, 0, stream>>>(batch, cnt, N_NODES);
  k_pool_final<<<cdivu((long long)N_GRAPH * F1, 256), B, 0, stream>>>(
      pool, cnt, d_out, N_GRAPH * F1, F1);

  // ---- layer 2: h2 = x_mid @ W2^T (20000x512 · 512x64), aggregate into d_out tail ----
  {
    int waves = (N_NODES / 16) * (HID / 64);          // 1250 * 1
    k_wmma_gemm_nt4<<<cdivu(waves, 8), B, 0, stream>>>(out1, W2, h2,
                                                       N_NODES / 16, HID / 64, F1, HID);
  }
  k_node_attn<<<cdivu(N_NODES, 256), B, 0, stream>>>(
      h2, att_src2, att_dst2, as2, ad2, N_NODES, 1, HID);
  k_edge_logit_max<<<cdivu(EPRIME, 256), B, 0, stream>>>(ei, as2, ad2, eatt, m2, 1);
  k_edge_exp_sum<<<cdivu(EPRIME, 256), B, 0, stream>>>(ei, eatt, m2, z2, 1);
  k_edge_scatter<<<cdivu((long long)EPRIME * (HID / 4), 256), B, 0, stream>>>(
      ei, eatt, z2, h2, xlate, 1, HID);
  k_add_bias<<<cdivu((long long)N_NODES * HID, 256), B, 0, stream>>>(
      xlate, b2, N_NODES * HID, HID);
}